// IndexedLinear_88768384074296
// MI455X (gfx1250) — compile-verified
//
#include <hip/hip_runtime.h>

// Indexed e3nn linear (64x0e + 32x1o + 16x2e), fp32, N=65536, 64 weight types.
// Bandwidth-bound: ~126 MB mandatory traffic -> ~5.4us floor @ 23.3 TB/s.
// Pipeline: bucket-by-type (count/prefix/scatter into 16-aligned, -1-padded
// slots) then three grouped-GEMM kernels using V_WMMA_F32_16X16X4_F32.

typedef __attribute__((ext_vector_type(2))) float v2f;
typedef __attribute__((ext_vector_type(8))) float v8f;

#define N_NODES   65536
#define NUM_TYPES 64
#define DIM       240
#define LIST_PAD  (N_NODES + NUM_TYPES * 16)   // worst-case 16-alignment padding

// workspace layout (int units)
#define WS_CNT  0      // 64 counters
#define WS_CUR  64     // 64 scatter cursors
#define WS_OFF  128    // 65 exclusive 16-aligned offsets (slot units)
#define WS_LIST 256    // LIST_PAD node ids (-1 = padding)

__global__ void il_init(int* __restrict__ ws) {
    int i = blockIdx.x * blockDim.x + threadIdx.x;
    if (i < 128) ws[WS_CNT + i] = 0;           // counters + cursors
    if (i < LIST_PAD) ws[WS_LIST + i] = -1;    // padded slots
}

__global__ void il_count(const int* __restrict__ idx, int* __restrict__ ws) {
    int n = blockIdx.x * blockDim.x + threadIdx.x;
    if (n < N_NODES) atomicAdd(&ws[WS_CNT + idx[n]], 1);
}

__global__ void il_prefix(int* __restrict__ ws) {
    if (threadIdx.x == 0) {
        int acc = 0;
        for (int t = 0; t < NUM_TYPES; ++t) {
            ws[WS_OFF + t] = acc;
            acc += (ws[WS_CNT + t] + 15) & ~15;   // 16-align each bucket
        }
        ws[WS_OFF + NUM_TYPES] = acc;
    }
}

__global__ void il_scatter(const int* __restrict__ idx, int* __restrict__ ws) {
    int n = blockIdx.x * blockDim.x + threadIdx.x;
    if (n < N_NODES) {
        int t = idx[n];
        int p = atomicAdd(&ws[WS_CUR + t], 1);
        ws[WS_LIST + ws[WS_OFF + t] + p] = n;
    }
}

__device__ __forceinline__ int find_type(const int* __restrict__ offs, int row) {
    int t = 0;
    while (t < NUM_TYPES - 1 && row >= offs[t + 1]) ++t;
    return t;
}

// ---------------- l=0 : 64x0e.  M=16 nodes, N=64, K=64 ----------------
__global__ __launch_bounds__(128) void il_block0(
    const float* __restrict__ x, const float* __restrict__ W0,
    const int* __restrict__ ws, float* __restrict__ out) {
    __shared__ float Xs[16 * 65];     // +1 pad: A reads stride-65 -> no bank conflict
    __shared__ float Ws[64 * 64];
    __shared__ int   nodeIds[16];

    const int* offs = ws + WS_OFF;
    const int* list = ws + WS_LIST;
    const int tileRow = blockIdx.x * 16;
    if (tileRow >= offs[NUM_TYPES]) return;                 // block-uniform exit
    const int t   = find_type(offs, tileRow);
    const int tid = threadIdx.x;

    if (tid < 16) nodeIds[tid] = list[tileRow + tid];
    __syncthreads();

    for (int j = tid; j < 16 * 64; j += 128) {
        int m = j >> 6, k = j & 63;
        int node = nodeIds[m];
        Xs[m * 65 + k] = (node >= 0) ? x[(size_t)node * DIM + k] : 0.0f;
    }
    const float* Wt = W0 + (size_t)t * 64 * 64;             // [i][o], o contiguous
    for (int j = tid; j < 64 * 64; j += 128) Ws[j] = Wt[j];
    __syncthreads();

    const int lane = tid & 31, w = tid >> 5;                // wave w: outputs [16w,16w+16)
    const int half = lane >> 4, l16 = lane & 15, kb = half * 2;

    v8f c = {0.f, 0.f, 0.f, 0.f, 0.f, 0.f, 0.f, 0.f};
#pragma unroll
    for (int kk = 0; kk < 16; ++kk) {
        int k = kk * 4 + kb;
        v2f a, b;
        a.x = Xs[l16 * 65 + k];
        a.y = Xs[l16 * 65 + k + 1];
        b.x = Ws[k * 64 + w * 16 + l16];
        b.y = Ws[(k + 1) * 64 + w * 16 + l16];
        c = __builtin_amdgcn_wmma_f32_16x16x4_f32(false, a, false, b,
                                                  (short)0, c, false, false);
    }
    const float scale = 0.125f;                             // 1/sqrt(64)
#pragma unroll
    for (int r = 0; r < 8; ++r) {
        int m = r + half * 8;
        int node = nodeIds[m];
        if (node >= 0) out[(size_t)node * DIM + w * 16 + l16] = c[r] * scale;
    }
}

// ---------------- l=1 : 32x1o.  rows=(node,d) 48 -> 3 M-tiles, N=32, K=32 ----
__global__ __launch_bounds__(192) void il_block1(
    const float* __restrict__ x, const float* __restrict__ W1,
    const int* __restrict__ ws, float* __restrict__ out) {
    __shared__ float Xs[16 * 97];     // contiguous copy of x[node,64:160], +1 pad
    __shared__ float Ws[32 * 32];
    __shared__ int   nodeIds[16];

    const int* offs = ws + WS_OFF;
    const int* list = ws + WS_LIST;
    const int tileRow = blockIdx.x * 16;
    if (tileRow >= offs[NUM_TYPES]) return;
    const int t   = find_type(offs, tileRow);
    const int tid = threadIdx.x;

    if (tid < 16) nodeIds[tid] = list[tileRow + tid];
    __syncthreads();

    for (int j = tid; j < 16 * 96; j += 192) {
        int m = j / 96, k = j - m * 96;
        int node = nodeIds[m];
        Xs[m * 97 + k] = (node >= 0) ? x[(size_t)node * DIM + 64 + k] : 0.0f;
    }
    const float* Wt = W1 + (size_t)t * 32 * 32;
    for (int j = tid; j < 32 * 32; j += 192) Ws[j] = Wt[j];
    __syncthreads();

    const int lane = tid & 31, w = tid >> 5;
    const int mt = w >> 1, nt = w & 1;                      // 3 M-tiles x 2 N-tiles
    const int half = lane >> 4, l16 = lane & 15, kb = half * 2;
    const int g = mt * 16 + l16;                            // A-row = (node,comp)
    const int slotA = g / 3, dcA = g - slotA * 3;

    v8f c = {0.f, 0.f, 0.f, 0.f, 0.f, 0.f, 0.f, 0.f};
#pragma unroll
    for (int kk = 0; kk < 8; ++kk) {
        int k = kk * 4 + kb;
        v2f a, b;
        a.x = Xs[slotA * 97 + k * 3 + dcA];
        a.y = Xs[slotA * 97 + (k + 1) * 3 + dcA];
        b.x = Ws[k * 32 + nt * 16 + l16];
        b.y = Ws[(k + 1) * 32 + nt * 16 + l16];
        c = __builtin_amdgcn_wmma_f32_16x16x4_f32(false, a, false, b,
                                                  (short)0, c, false, false);
    }
    const float scale = 0.17677669529663687f;               // 1/sqrt(32)
#pragma unroll
    for (int r = 0; r < 8; ++r) {
        int grow = mt * 16 + r + half * 8;
        int slot = grow / 3, dc = grow - slot * 3;
        int node = nodeIds[slot];
        int o = nt * 16 + l16;
        if (node >= 0)
            out[(size_t)node * DIM + 64 + o * 3 + dc] = c[r] * scale;
    }
}

// ---------------- l=2 : 16x2e.  rows=(node,d) 80 -> 5 M-tiles, N=16, K=16 ----
__global__ __launch_bounds__(160) void il_block2(
    const float* __restrict__ x, const float* __restrict__ W2,
    const int* __restrict__ ws, float* __restrict__ out) {
    __shared__ float Xs[16 * 81];     // contiguous copy of x[node,160:240], +1 pad
    __shared__ float Ws[16 * 16];
    __shared__ int   nodeIds[16];

    const int* offs = ws + WS_OFF;
    const int* list = ws + WS_LIST;
    const int tileRow = blockIdx.x * 16;
    if (tileRow >= offs[NUM_TYPES]) return;
    const int t   = find_type(offs, tileRow);
    const int tid = threadIdx.x;

    if (tid < 16) nodeIds[tid] = list[tileRow + tid];
    __syncthreads();

    for (int j = tid; j < 16 * 80; j += 160) {
        int m = j / 80, k = j - m * 80;
        int node = nodeIds[m];
        Xs[m * 81 + k] = (node >= 0) ? x[(size_t)node * DIM + 160 + k] : 0.0f;
    }
    const float* Wt = W2 + (size_t)t * 16 * 16;
    for (int j = tid; j < 16 * 16; j += 160) Ws[j] = Wt[j];
    __syncthreads();

    const int lane = tid & 31, mt = tid >> 5;               // 5 M-tiles, single N-tile
    const int half = lane >> 4, l16 = lane & 15, kb = half * 2;
    const int g = mt * 16 + l16;
    const int slotA = g / 5, dcA = g - slotA * 5;

    v8f c = {0.f, 0.f, 0.f, 0.f, 0.f, 0.f, 0.f, 0.f};
#pragma unroll
    for (int kk = 0; kk < 4; ++kk) {
        int k = kk * 4 + kb;
        v2f a, b;
        a.x = Xs[slotA * 81 + k * 5 + dcA];
        a.y = Xs[slotA * 81 + (k + 1) * 5 + dcA];
        b.x = Ws[k * 16 + l16];
        b.y = Ws[(k + 1) * 16 + l16];
        c = __builtin_amdgcn_wmma_f32_16x16x4_f32(false, a, false, b,
                                                  (short)0, c, false, false);
    }
    const float scale = 0.25f;                              // 1/sqrt(16)
#pragma unroll
    for (int r = 0; r < 8; ++r) {
        int grow = mt * 16 + r + half * 8;
        int slot = grow / 5, dc = grow - slot * 5;
        int node = nodeIds[slot];
        if (node >= 0)
            out[(size_t)node * DIM + 160 + l16 * 5 + dc] = c[r] * scale;
    }
}

extern "C" void kernel_launch(void* const* d_in, const int* in_sizes, int n_in,
                              void* d_out, int out_size, void* d_ws, size_t ws_size,
                              hipStream_t stream) {
    (void)in_sizes; (void)n_in; (void)out_size; (void)ws_size;
    const float* x  = (const float*)d_in[0];
    const float* W0 = (const float*)d_in[1];
    const float* W1 = (const float*)d_in[2];
    const float* W2 = (const float*)d_in[3];
    const int*   idx = (const int*)d_in[4];
    float* out = (float*)d_out;
    int*   ws  = (int*)d_ws;   // needs ~261 KB

    il_init   <<<(LIST_PAD + 255) / 256, 256, 0, stream>>>(ws);
    il_count  <<<(N_NODES + 255) / 256, 256, 0, stream>>>(idx, ws);
    il_prefix <<<1, 64, 0, stream>>>(ws);
    il_scatter<<<(N_NODES + 255) / 256, 256, 0, stream>>>(idx, ws);

    const int maxTiles = LIST_PAD / 16;   // 4160 (worst case, empty tiles exit fast)
    il_block0<<<maxTiles, 128, 0, stream>>>(x, W0, ws, out);
    il_block1<<<maxTiles, 192, 0, stream>>>(x, W1, ws, out);
    il_block2<<<maxTiles, 160, 0, stream>>>(x, W2, ws, out);
}